// Attention_40724879901066
// MI455X (gfx1250) — compile-verified
//
#include <hip/hip_runtime.h>
#include <math.h>
#include <stdint.h>

// Problem constants (match reference)
#define BDIM 4
#define HDIM 16
#define SEQ 4096
#define DDIM 64
#define MDIM 64
#define DVDIM 64
#define CHUNKC 128
#define NCHUNK (SEQ / CHUNKC)   // 32
#define BH (BDIM * HDIM)        // 64
#define FEPS 1e-4f
#define DNORM 0.35355339059327379f  // 64^-0.25
#define DGSC  0.0625f               // 0.5 * DNORM^2 = 0.5 * 64^-0.5
#define RATIO 0.125f                // 64^-0.5

typedef __attribute__((ext_vector_type(2))) float v2f;
typedef __attribute__((ext_vector_type(8))) float v8f;
typedef int v4i __attribute__((ext_vector_type(4)));

// ---------------------------------------------------------------------------
// Async global -> LDS staging (CDNA5 GLOBAL_LOAD_ASYNC_TO_LDS_B128, ASYNCcnt).
// Builtin signature (from clang diagnostic): param0 is `int __vector(4) *`
// (generic); LDS side passed as addrspace(3) vector pointer.
// Guarded: falls back to a VGPR round-trip copy if the builtin is absent.
// ---------------------------------------------------------------------------
#define ASYNC_OK __has_builtin(__builtin_amdgcn_global_load_async_to_lds_b128)

__device__ __forceinline__ void copy_b128(float* __restrict__ dst_lds,
                                          const float* __restrict__ src) {
#if ASYNC_OK
  typedef __attribute__((address_space(3))) v4i* lds_v4i_p;
  v4i* g = (v4i*)(uintptr_t)src;                         // generic global ptr
  lds_v4i_p l = (lds_v4i_p)(unsigned)(uintptr_t)dst_lds; // low 32b = LDS offset
  __builtin_amdgcn_global_load_async_to_lds_b128(g, l, 0, 0);
#else
  *(float4*)dst_lds = *(const float4*)src;
#endif
}

__device__ __forceinline__ void async_wait_all() {
#if ASYNC_OK
#if __has_builtin(__builtin_amdgcn_s_wait_asynccnt)
  __builtin_amdgcn_s_wait_asynccnt(0);
#else
  asm volatile("s_wait_asynccnt 0x0" ::: "memory");
#endif
#endif
}

// ---------------------------------------------------------------------------
// fp32 WMMA helpers: D(16x16) += A(16x4) * B(4x16), K-stepped by 4.
// Fragment layout per CDNA5 ISA 7.12.2:
//   A 16x4 f32:  lane l (0-15): M=l, VGPR0=K0,VGPR1=K1 ; lanes 16-31: K2,K3
//   B 4x16 f32:  VGPR0: lanes0-15 K=0 / lanes16-31 K=2 ; VGPR1: K=1 / K=3
//   C/D 16x16:   VGPR v: lanes0-15 -> M=v, lanes16-31 -> M=v+8 ; N=lane%16
// ---------------------------------------------------------------------------
__device__ __forceinline__ v8f wmma_k4(v2f a, v2f b, v8f c) {
  return __builtin_amdgcn_wmma_f32_16x16x4_f32(
      false, a, false, b, (short)0, c, false, false);
}

// C(16x16) += A(16xK) * B(Kx16); A row-major (lda), B row-major (ldb).
__device__ __forceinline__ v8f mma_tile(const float* __restrict__ A, int lda,
                                        const float* __restrict__ B, int ldb,
                                        int klen, v8f acc) {
  const int lane = threadIdx.x & 31;
  const int half = lane >> 4;
  const int lid  = lane & 15;
  for (int kk = 0; kk < klen; kk += 4) {
    v2f a, b;
    a.x = A[lid * lda + kk + 2 * half];
    a.y = A[lid * lda + kk + 2 * half + 1];
    b.x = B[(kk + 2 * half) * ldb + lid];
    b.y = B[(kk + 2 * half + 1) * ldb + lid];
    acc = wmma_k4(a, b, acc);
  }
  return acc;
}

// C(16x16) += A^T * B, where At is stored K-major: A[m][k] = At[k*ldat + m].
__device__ __forceinline__ v8f mma_tile_At(const float* __restrict__ At, int ldat,
                                           const float* __restrict__ B, int ldb,
                                           int klen, v8f acc) {
  const int lane = threadIdx.x & 31;
  const int half = lane >> 4;
  const int lid  = lane & 15;
  for (int kk = 0; kk < klen; kk += 4) {
    v2f a, b;
    a.x = At[(kk + 2 * half) * ldat + lid];
    a.y = At[(kk + 2 * half + 1) * ldat + lid];
    b.x = B[(kk + 2 * half) * ldb + lid];
    b.y = B[(kk + 2 * half + 1) * ldb + lid];
    acc = wmma_k4(a, b, acc);
  }
  return acc;
}

// C(16x16) += A * B^T, where Bt is stored N-major: B[k][n] = Bt[n*ldbt + k].
__device__ __forceinline__ v8f mma_tile_Bt(const float* __restrict__ A, int lda,
                                           const float* __restrict__ Bt, int ldbt,
                                           int klen, v8f acc) {
  const int lane = threadIdx.x & 31;
  const int half = lane >> 4;
  const int lid  = lane & 15;
  for (int kk = 0; kk < klen; kk += 4) {
    v2f a, b;
    a.x = A[lid * lda + kk + 2 * half];
    a.y = A[lid * lda + kk + 2 * half + 1];
    b.x = Bt[lid * ldbt + kk + 2 * half];
    b.y = Bt[lid * ldbt + kk + 2 * half + 1];
    acc = wmma_k4(a, b, acc);
  }
  return acc;
}

__device__ __forceinline__ void store_tile(float* __restrict__ C, int ldc, v8f acc) {
  const int lane = threadIdx.x & 31;
  const int half = lane >> 4;
  const int lid  = lane & 15;
#pragma unroll
  for (int vv = 0; vv < 8; ++vv)
    C[(vv + 8 * half) * ldc + lid] = acc[vv];
}

// ---------------------------------------------------------------------------
// Kernel 1: FAVOR+ feature map for one (b,h,chunk) tile of 128 rows.
//   dd = DNORM * (x @ proj)   (WMMA on raw x, scale at consumption)
//   is_query: feat = RATIO*(exp(dd - diag - rowmax(dd)) + EPS)
//   else    : feat = dd - diag (raw), blockmax[blk] = max(dd) for global stab
// ---------------------------------------------------------------------------
extern "C" __global__ __launch_bounds__(256)
void feat_kernel(const float* __restrict__ x, const float* __restrict__ proj,
                 float* __restrict__ feat, float* __restrict__ blockmax,
                 int is_query) {
  extern __shared__ float smem[];
  float* sx    = smem;                   // 128*64 (raw x)
  float* sp    = sx + CHUNKC * DDIM;     // 64*64
  float* sdd   = sp + DDIM * MDIM;       // 128*64 (x @ proj, unscaled)
  float* sdiag = sdd + CHUNKC * MDIM;    // 128
  float* sred  = sdiag + CHUNKC;         // 128

  const int tid = threadIdx.x;
  const int bhc = blockIdx.x;
  const int bh  = bhc / NCHUNK;
  const int ch  = bhc % NCHUNK;
  const long base = ((long)bh * SEQ + (long)ch * CHUNKC) * DDIM;

  for (int i = tid; i < CHUNKC * DDIM / 4; i += 256)
    copy_b128(sx + 4 * i, x + base + 4 * i);
  for (int i = tid; i < DDIM * MDIM / 4; i += 256)
    copy_b128(sp + 4 * i, proj + 4 * i);
  async_wait_all();
  __syncthreads();

  // diag = 0.5 * dnorm^2 * sum(x^2)
  if (tid < CHUNKC) {
    const float* row = sx + tid * DDIM;
    float s = 0.f;
#pragma unroll 8
    for (int j = 0; j < DDIM; ++j) s += row[j] * row[j];
    sdiag[tid] = DGSC * s;
  }

  // 8 waves: wave w -> row tile w (16 rows), 4 column tiles of 16
  const int w = tid >> 5;
  for (int ct = 0; ct < 4; ++ct) {
    v8f acc = {};
    acc = mma_tile(sx + (w * 16) * DDIM, DDIM, sp + ct * 16, MDIM, DDIM, acc);
    store_tile(sdd + (w * 16) * MDIM + ct * 16, MDIM, acc);
  }
  __syncthreads();

  if (tid < CHUNKC) {
    const float* row = sdd + tid * MDIM;
    float mx = DNORM * row[0];
    for (int j = 1; j < MDIM; ++j) mx = fmaxf(mx, DNORM * row[j]);
    const long ob = ((long)bh * SEQ + (long)ch * CHUNKC + tid) * MDIM;
    const float dg = sdiag[tid];
    if (is_query) {
      for (int j = 0; j < MDIM; ++j)
        feat[ob + j] = RATIO * (expf(DNORM * row[j] - dg - mx) + FEPS);
    } else {
      for (int j = 0; j < MDIM; ++j)
        feat[ob + j] = DNORM * row[j] - dg;  // finished later with global stab
      sred[tid] = mx;                        // stab uses max of raw data_dash
    }
  }
  if (!is_query) {
    __syncthreads();
    if (tid == 0) {
      float mx = sred[0];
      for (int j = 1; j < CHUNKC; ++j) mx = fmaxf(mx, sred[j]);
      blockmax[bhc] = mx;
    }
  }
}

// Kernel 2: global max over block maxima -> gmax[0]
extern "C" __global__ __launch_bounds__(256)
void reduce_max_kernel(const float* __restrict__ bmax, float* __restrict__ gmax,
                       int n) {
  __shared__ float sred[256];
  float mx = -3.4e38f;
  for (int i = threadIdx.x; i < n; i += 256) mx = fmaxf(mx, bmax[i]);
  sred[threadIdx.x] = mx;
  __syncthreads();
  for (int s = 128; s > 0; s >>= 1) {
    if (threadIdx.x < s)
      sred[threadIdx.x] = fmaxf(sred[threadIdx.x], sred[threadIdx.x + s]);
    __syncthreads();
  }
  if (threadIdx.x == 0) gmax[0] = sred[0];
}

// Kernel 3: kp = RATIO*(exp(raw - gmax) + EPS), in place
extern "C" __global__ __launch_bounds__(256)
void finalize_kp_kernel(float* __restrict__ kp, const float* __restrict__ gmax,
                        long n) {
  const float gm = gmax[0];
  long i = (long)blockIdx.x * blockDim.x + threadIdx.x;
  const long stride = (long)gridDim.x * blockDim.x;
  for (; i < n; i += stride) kp[i] = RATIO * (expf(kp[i] - gm) + FEPS);
}

// Kernel 4: per-chunk sums  ckv = kp^T @ v (64x64, K=128), cks = colsum(kp)
extern "C" __global__ __launch_bounds__(256)
void chunk_sums_kernel(const float* __restrict__ kp, const float* __restrict__ v,
                       float* __restrict__ ckv, float* __restrict__ cks) {
  extern __shared__ float smem[];
  float* skp = smem;                    // 128*64
  float* sv  = skp + CHUNKC * MDIM;     // 128*64

  const int tid = threadIdx.x;
  const int bhc = blockIdx.x;
  const int bh  = bhc / NCHUNK;
  const int ch  = bhc % NCHUNK;
  const long kb = ((long)bh * SEQ + (long)ch * CHUNKC) * MDIM;
  const long vb = ((long)bh * SEQ + (long)ch * CHUNKC) * DVDIM;

  for (int i = tid; i < CHUNKC * MDIM / 4; i += 256) {
    copy_b128(skp + 4 * i, kp + kb + 4 * i);
    copy_b128(sv + 4 * i, v + vb + 4 * i);
  }
  async_wait_all();
  __syncthreads();

  if (tid < MDIM) {
    float s = 0.f;
    for (int c = 0; c < CHUNKC; ++c) s += skp[c * MDIM + tid];
    cks[(long)bhc * MDIM + tid] = s;
  }

  const int w = tid >> 5;               // 8 waves x 2 tiles = 16 tiles (4x4)
  for (int i = 0; i < 2; ++i) {
    const int t = w * 2 + i, tr = t >> 2, tc = t & 3;
    v8f acc = {};
    acc = mma_tile_At(skp + tr * 16, MDIM, sv + tc * 16, DVDIM, CHUNKC, acc);
    store_tile(ckv + (long)bhc * MDIM * DVDIM + (tr * 16) * DVDIM + tc * 16,
               DVDIM, acc);
  }
}

// Kernel 5: exclusive prefix over chunks (in place) -> running kv/ks state
extern "C" __global__ __launch_bounds__(256)
void prefix_kernel(float* __restrict__ ckv, float* __restrict__ cks) {
  const int bh = blockIdx.x, tid = threadIdx.x;
  for (int e = tid; e < MDIM * DVDIM; e += 256) {
    float run = 0.f;
    for (int ch = 0; ch < NCHUNK; ++ch) {
      const long idx = ((long)bh * NCHUNK + ch) * MDIM * DVDIM + e;
      if (ch + 1 < NCHUNK)
        __builtin_prefetch(&ckv[idx + MDIM * DVDIM], 1, 0);  // global_prefetch
      const float val = ckv[idx];
      ckv[idx] = run;
      run += val;
    }
  }
  if (tid < MDIM) {
    float run = 0.f;
    for (int ch = 0; ch < NCHUNK; ++ch) {
      const long idx = ((long)bh * NCHUNK + ch) * MDIM + tid;
      const float val = cks[idx];
      cks[idx] = run;
      run += val;
    }
  }
}

// Kernel 6: per-chunk output.
//   A   = qp @ kp^T  (128x128, masked tril)            [WMMA]
//   num = A @ v + qp @ kv_prefix                       [WMMA]
//   den = rowsum(A) + qp . ks_prefix
//   out = num / den
extern "C" __global__ __launch_bounds__(256)
void out_kernel(const float* __restrict__ qp, const float* __restrict__ kp,
                const float* __restrict__ v, const float* __restrict__ ckv,
                const float* __restrict__ cks, float* __restrict__ out) {
  extern __shared__ float smem[];
  float* sq   = smem;                     // 128*64
  float* sk   = sq + CHUNKC * MDIM;       // 128*64
  float* sv   = sk + CHUNKC * MDIM;       // 128*64
  float* skv  = sv + CHUNKC * DVDIM;      // 64*64
  float* sks  = skv + MDIM * DVDIM;       // 64
  float* sden = sks + MDIM;               // 128
  float* sA   = sden + CHUNKC;            // 128*128

  const int tid = threadIdx.x;
  const int bhc = blockIdx.x;
  const int bh  = bhc / NCHUNK;
  const int ch  = bhc % NCHUNK;
  const long fb = ((long)bh * SEQ + (long)ch * CHUNKC) * MDIM;
  const long vb = ((long)bh * SEQ + (long)ch * CHUNKC) * DVDIM;

  for (int i = tid; i < CHUNKC * MDIM / 4; i += 256) {
    copy_b128(sq + 4 * i, qp + fb + 4 * i);
    copy_b128(sk + 4 * i, kp + fb + 4 * i);
    copy_b128(sv + 4 * i, v + vb + 4 * i);
  }
  for (int i = tid; i < MDIM * DVDIM / 4; i += 256)
    copy_b128(skv + 4 * i, ckv + (long)bhc * MDIM * DVDIM + 4 * i);
  if (tid < MDIM) sks[tid] = cks[(long)bhc * MDIM + tid];
  async_wait_all();
  __syncthreads();

  const int w = tid >> 5;
  const int lane = tid & 31, half = lane >> 4, lid = lane & 15;

  // A strip: wave w -> rows [16w,16w+16), all 8 column tiles; causal mask.
  for (int tc = 0; tc < 8; ++tc) {
    v8f acc = {};
    acc = mma_tile_Bt(sq + (w * 16) * MDIM, MDIM, sk + (tc * 16) * MDIM, MDIM,
                      MDIM, acc);
#pragma unroll
    for (int vv = 0; vv < 8; ++vv) {
      const int R = w * 16 + vv + 8 * half;
      const int Ccol = tc * 16 + lid;
      sA[R * CHUNKC + Ccol] = (Ccol <= R) ? acc[vv] : 0.f;
    }
  }
  __syncthreads();

  if (tid < CHUNKC) {
    const float* row = sA + tid * CHUNKC;
    float s = 0.f;
    for (int j = 0; j < CHUNKC; ++j) s += row[j];
    const float* qrow = sq + tid * MDIM;
    for (int j = 0; j < MDIM; ++j) s += qrow[j] * sks[j];
    sden[tid] = s;
  }
  __syncthreads();

  for (int tc = 0; tc < 4; ++tc) {
    v8f acc = {};
    acc = mma_tile(sA + (w * 16) * CHUNKC, CHUNKC, sv + tc * 16, DVDIM, CHUNKC,
                   acc);
    acc = mma_tile(sq + (w * 16) * MDIM, MDIM, skv + tc * 16, DVDIM, MDIM, acc);
#pragma unroll
    for (int vv = 0; vv < 8; ++vv) {
      const int R = w * 16 + vv + 8 * half;
      out[vb + (long)R * DVDIM + tc * 16 + lid] = acc[vv] / sden[R];
    }
  }
}

// ---------------------------------------------------------------------------
extern "C" void kernel_launch(void* const* d_in, const int* in_sizes, int n_in,
                              void* d_out, int out_size, void* d_ws,
                              size_t ws_size, hipStream_t stream) {
  const float* q    = (const float*)d_in[0];
  const float* k    = (const float*)d_in[1];
  const float* v    = (const float*)d_in[2];
  const float* proj = (const float*)d_in[3];
  float* out = (float*)d_out;

  float* ws = (float*)d_ws;
  const long FEAT = (long)BH * SEQ * MDIM;            // 16,777,216 floats
  float* qp   = ws;
  float* kp   = qp + FEAT;
  float* ckv  = kp + FEAT;                            // BH*NC*64*64
  float* cks  = ckv + (long)BH * NCHUNK * MDIM * DVDIM;
  float* bmax = cks + (long)BH * NCHUNK * MDIM;
  float* gmax = bmax + BH * NCHUNK;

  const size_t featShm =
      (size_t)(CHUNKC * DDIM + DDIM * MDIM + CHUNKC * MDIM + 2 * CHUNKC) *
      sizeof(float);
  const size_t csumShm = (size_t)(CHUNKC * MDIM + CHUNKC * DVDIM) * sizeof(float);
  const size_t outShm =
      (size_t)(2 * CHUNKC * MDIM + CHUNKC * DVDIM + MDIM * DVDIM + MDIM +
               CHUNKC + CHUNKC * CHUNKC) * sizeof(float);

  (void)hipFuncSetAttribute(reinterpret_cast<const void*>(feat_kernel),
                            hipFuncAttributeMaxDynamicSharedMemorySize,
                            (int)featShm);
  (void)hipFuncSetAttribute(reinterpret_cast<const void*>(chunk_sums_kernel),
                            hipFuncAttributeMaxDynamicSharedMemorySize,
                            (int)csumShm);
  (void)hipFuncSetAttribute(reinterpret_cast<const void*>(out_kernel),
                            hipFuncAttributeMaxDynamicSharedMemorySize,
                            (int)outShm);

  feat_kernel<<<BH * NCHUNK, 256, featShm, stream>>>(q, proj, qp, bmax, 1);
  feat_kernel<<<BH * NCHUNK, 256, featShm, stream>>>(k, proj, kp, bmax, 0);
  reduce_max_kernel<<<1, 256, 0, stream>>>(bmax, gmax, BH * NCHUNK);
  finalize_kp_kernel<<<2048, 256, 0, stream>>>(kp, gmax, FEAT);
  chunk_sums_kernel<<<BH * NCHUNK, 256, csumShm, stream>>>(kp, v, ckv, cks);
  prefix_kernel<<<BH, 256, 0, stream>>>(ckv, cks);
  out_kernel<<<BH * NCHUNK, 256, outShm, stream>>>(qp, kp, v, ckv, cks, out);
}